// GaussianKernel_14370960572473
// MI455X (gfx1250) — compile-verified
//
#include <hip/hip_runtime.h>
#include <math.h>

// GaussianKernel on MI455X (gfx1250, wave32, WMMA):
//   phase 1: sq[i] = ||x_i||^2, zero the global d2-sum accumulator
//   phase 2: WMMA f32 16x16x4 GEMM -> d2 tiles -> store d2 to d_out,
//            wave-reduced partial sums -> one atomicAdd per wave
//   phase 3: out = exp(-d2 * N^2 / (2*sum))  (in place over d_out)

typedef float v2f __attribute__((ext_vector_type(2)));
typedef float v8f __attribute__((ext_vector_type(8)));

#define NROWS 8192
#define KDIM  64
#define REG   32                       // 32x32 output region per wave
#define REGIONS_PER_ROW (NROWS / REG)  // 256
#define WAVES_PER_BLOCK 8
#define TPB (WAVES_PER_BLOCK * 32)

// ---------------- phase 1: row squared-norms + zero the accumulator --------
__global__ void sqinit_kernel(const float* __restrict__ X,
                              float* __restrict__ sq,
                              float* __restrict__ sumAcc) {
  int r = blockIdx.x * blockDim.x + threadIdx.x;
  if (r == 0) *sumAcc = 0.0f;
  if (r < NROWS) {
    const float4* xr = (const float4*)(X + (size_t)r * KDIM);
    float s = 0.0f;
#pragma unroll
    for (int i = 0; i < KDIM / 4; ++i) {
      float4 v = xr[i];
      s += v.x * v.x + v.y * v.y + v.z * v.z + v.w * v.w;
    }
    sq[r] = s;
  }
}

// ---------------- phase 2: WMMA GEMM -> d2 + global sum --------------------
__global__ __launch_bounds__(TPB) void d2_wmma_kernel(
    const float* __restrict__ X, const float* __restrict__ sq,
    float* __restrict__ d2out, float* __restrict__ sumAcc) {
  const int lane = threadIdx.x & 31;
  const int wave = threadIdx.x >> 5;
  const int region = blockIdx.x * WAVES_PER_BLOCK + wave;
  const int rm = (region / REGIONS_PER_ROW) * REG;   // row-tile base
  const int rn = (region % REGIONS_PER_ROW) * REG;   // col-tile base
  const int half = lane >> 4;                        // 0: lanes 0-15, 1: 16-31
  const int l16  = lane & 15;

  // A fragment: M = l16 (+16 for second row tile), K pair = 2*half + {0,1}
  // B fragment of X*X^T is X read identically with the column-tile base.
  const float* arow0 = X + (size_t)(rm + l16) * KDIM + 2 * half;
  const float* arow1 = X + (size_t)(rm + 16 + l16) * KDIM + 2 * half;
  const float* brow0 = X + (size_t)(rn + l16) * KDIM + 2 * half;
  const float* brow1 = X + (size_t)(rn + 16 + l16) * KDIM + 2 * half;

  v8f c00 = {}; v8f c01 = {}; v8f c10 = {}; v8f c11 = {};

#pragma unroll
  for (int k0 = 0; k0 < KDIM; k0 += 4) {
    v2f a0 = *(const v2f*)(arow0 + k0);
    v2f a1 = *(const v2f*)(arow1 + k0);
    v2f b0 = *(const v2f*)(brow0 + k0);
    v2f b1 = *(const v2f*)(brow1 + k0);
    // 8 args: (neg_a, A, neg_b, B, c_mod, C, reuse_a, reuse_b)
    c00 = __builtin_amdgcn_wmma_f32_16x16x4_f32(false, a0, false, b0, (short)0, c00, false, false);
    c01 = __builtin_amdgcn_wmma_f32_16x16x4_f32(false, a0, false, b1, (short)0, c01, false, false);
    c10 = __builtin_amdgcn_wmma_f32_16x16x4_f32(false, a1, false, b0, (short)0, c10, false, false);
    c11 = __builtin_amdgcn_wmma_f32_16x16x4_f32(false, a1, false, b1, (short)0, c11, false, false);
  }

  // C/D layout: VGPR r -> M = r + 8*half, N = l16
  const float sqn0 = sq[rn + l16];
  const float sqn1 = sq[rn + 16 + l16];
  float lsum = 0.0f;

#pragma unroll
  for (int r = 0; r < 8; ++r) {
    const int m0 = rm + r + 8 * half;
    const int m1 = m0 + 16;
    const float sqm0 = sq[m0];
    const float sqm1 = sq[m1];
    float d;
    d = fmaxf(sqm0 + sqn0 - 2.0f * c00[r], 0.0f);
    d2out[(size_t)m0 * NROWS + rn + l16] = d;      lsum += d;
    d = fmaxf(sqm0 + sqn1 - 2.0f * c01[r], 0.0f);
    d2out[(size_t)m0 * NROWS + rn + 16 + l16] = d; lsum += d;
    d = fmaxf(sqm1 + sqn0 - 2.0f * c10[r], 0.0f);
    d2out[(size_t)m1 * NROWS + rn + l16] = d;      lsum += d;
    d = fmaxf(sqm1 + sqn1 - 2.0f * c11[r], 0.0f);
    d2out[(size_t)m1 * NROWS + rn + 16 + l16] = d; lsum += d;
  }

  // wave32 butterfly reduction, then one atomic per wave
#pragma unroll
  for (int off = 16; off > 0; off >>= 1)
    lsum += __shfl_xor(lsum, off, 32);
  if (lane == 0) atomicAdd(sumAcc, lsum);
}

// ---------------- phase 3: exp(-d2 / (2 sigma^2)) in place -----------------
__global__ void finalize_kernel(float* __restrict__ out,
                                const float* __restrict__ sumAcc) {
  // sigma^2 = sum / N^2  =>  -1/(2 sigma^2) = -N^2 / (2*sum)
  const float s = *sumAcc;
  const float scale = -((float)NROWS * (float)NROWS) / (2.0f * s);
  const size_t idx = ((size_t)blockIdx.x * blockDim.x + threadIdx.x) * 4;
  float4 v = *(float4*)(out + idx);
  v.x = __expf(v.x * scale);
  v.y = __expf(v.y * scale);
  v.z = __expf(v.z * scale);
  v.w = __expf(v.w * scale);
  *(float4*)(out + idx) = v;
}

extern "C" void kernel_launch(void* const* d_in, const int* in_sizes, int n_in,
                              void* d_out, int out_size, void* d_ws, size_t ws_size,
                              hipStream_t stream) {
  (void)in_sizes; (void)n_in; (void)out_size; (void)ws_size;
  const float* X   = (const float*)d_in[0];
  float*       out = (float*)d_out;
  float*       sum = (float*)d_ws;          // [0]: global d2 sum
  float*       sq  = (float*)d_ws + 16;     // [16..16+8191]: row norms

  sqinit_kernel<<<(NROWS + 255) / 256, 256, 0, stream>>>(X, sq, sum);

  const int nRegions = REGIONS_PER_ROW * REGIONS_PER_ROW;        // 65536
  d2_wmma_kernel<<<nRegions / WAVES_PER_BLOCK, TPB, 0, stream>>>(X, sq, out, sum);

  const size_t nElems = (size_t)NROWS * NROWS;                   // 64M
  finalize_kernel<<<(unsigned)(nElems / 4 / 256), 256, 0, stream>>>(out, sum);
}